// Depthwise1d_5609227288828
// MI455X (gfx1250) — compile-verified
//
#include <hip/hip_runtime.h>

typedef __attribute__((ext_vector_type(2))) float v2f;
typedef __attribute__((ext_vector_type(4))) float v4f;
typedef __attribute__((ext_vector_type(8))) float v8f;

#define N_TOT   4096
#define C_TOT   256
#define H_IN    64
#define H_OUT   128
#define M_BLOCK 256          // rows of n per block (8 waves x 2 M-tiles x 16)
#define LDSW    68           // padded row stride (floats): banks step by 4 per row

__global__ __launch_bounds__(256) void dw1d_wmma_f32(
    const float* __restrict__ x, const float* __restrict__ W,
    const float* __restrict__ b, float* __restrict__ out)
{
    __shared__ float Wlds[H_OUT   * LDSW];  // 128*68*4 = 34816 B
    __shared__ float Xlds[M_BLOCK * LDSW];  // 256*68*4 = 69632 B

    const int tid    = threadIdx.x;
    const int lane   = tid & 31;
    const int wave   = tid >> 5;            // 0..7
    const int c      = blockIdx.y;
    const int m_base = blockIdx.x * M_BLOCK;

    // ---- stage W_c (128x64 fp32) into LDS, coalesced float4 ----
    const float* Wc = W + (size_t)c * H_OUT * H_IN;
    #pragma unroll
    for (int j = 0; j < 8; ++j) {
        int flat = tid + j * 256;           // float4 index, 2048 total
        int o    = flat >> 4;               // row (0..127)
        int i4   = (flat & 15) << 2;        // col (0..60 step 4)
        v4f v = *(const v4f*)(Wc + (size_t)o * H_IN + i4);
        *(v4f*)(&Wlds[o * LDSW + i4]) = v;
    }
    // ---- stage X tile (256 rows x 64) into LDS; 16 lanes cover one 256B row ----
    #pragma unroll
    for (int j = 0; j < 16; ++j) {
        int flat = tid + j * 256;
        int r    = flat >> 4;               // local row (0..255)
        int i4   = (flat & 15) << 2;
        const float* xr = x + ((size_t)(m_base + r) * C_TOT + c) * H_IN;
        v4f v = *(const v4f*)(xr + i4);
        *(v4f*)(&Xlds[r * LDSW + i4]) = v;
    }
    __syncthreads();

    // ---- accumulators init from bias: C/D element (M=r+8*(lane>=16), N=lane&15) ----
    const int ncol = lane & 15;
    const int koff = (lane >> 4) << 1;      // A/B lane-half K offset: 0 or 2
    v8f acc0[8], acc1[8];
    #pragma unroll
    for (int t = 0; t < 8; ++t) {
        float bv = b[c * H_OUT + t * 16 + ncol];
        acc0[t] = (v8f){bv, bv, bv, bv, bv, bv, bv, bv};
        acc1[t] = acc0[t];
    }

    // A fragment sources: this wave owns rows [wave*32, wave*32+32):
    //   tile0 = rows wave*32 + 0..15, tile1 = rows wave*32 + 16..31
    // A lane L holds row M=L&15; VGPR0=K(k+koff), VGPR1=K(k+koff+1) (16x4 fp32 A layout)
    const float* Arow0 = &Xlds[(wave * 32 +      ncol) * LDSW];
    const float* Arow1 = &Xlds[(wave * 32 + 16 + ncol) * LDSW];

    #pragma unroll
    for (int kb = 0; kb < 16; ++kb) {       // K = 64 -> 16 steps of K=4
        const int k = kb * 4 + koff;
        v2f a0 = *(const v2f*)(Arow0 + k);  // 8B-aligned (k even, row stride 272B)
        v2f a1 = *(const v2f*)(Arow1 + k);
        #pragma unroll
        for (int t = 0; t < 8; ++t) {       // 8 N-tiles of 16 -> H_OUT=128
            v2f bf = *(const v2f*)(&Wlds[(t * 16 + ncol) * LDSW + k]);
            acc0[t] = __builtin_amdgcn_wmma_f32_16x16x4_f32(
                false, a0, false, bf, (short)0, acc0[t], false, false);
            acc1[t] = __builtin_amdgcn_wmma_f32_16x16x4_f32(
                false, a1, false, bf, (short)0, acc1[t], false, false);
        }
    }

    // ---- store: n = m_base + wave*32 + mt*16 + r + 8*(lane>=16), o = t*16 + ncol ----
    const int rb = (lane >> 4) << 3;        // 0 or 8
    #pragma unroll
    for (int r = 0; r < 8; ++r) {
        #pragma unroll
        for (int t = 0; t < 8; ++t) {       // t-inner: adjacent 64B half-lines back-to-back
            int n0 = m_base + wave * 32 + rb + r;
            out[((size_t)n0 * C_TOT + c) * H_OUT + t * 16 + ncol] = acc0[t][r];
        }
    }
    #pragma unroll
    for (int r = 0; r < 8; ++r) {
        #pragma unroll
        for (int t = 0; t < 8; ++t) {
            int n1 = m_base + wave * 32 + 16 + rb + r;
            out[((size_t)n1 * C_TOT + c) * H_OUT + t * 16 + ncol] = acc1[t][r];
        }
    }
}

extern "C" void kernel_launch(void* const* d_in, const int* in_sizes, int n_in,
                              void* d_out, int out_size, void* d_ws, size_t ws_size,
                              hipStream_t stream) {
    const float* x = (const float*)d_in[0];
    const float* W = (const float*)d_in[1];
    const float* b = (const float*)d_in[2];
    float* out     = (float*)d_out;
    dim3 grid(N_TOT / M_BLOCK, C_TOT);      // (16, 256) blocks, 256 threads (8 wave32)
    dw1d_wmma_f32<<<grid, 256, 0, stream>>>(x, W, b, out);
}